// MultivariateMOLAutoregressiveSubsetTransform2d_69217692942842
// MI455X (gfx1250) — compile-verified
//
#include <hip/hip_runtime.h>

// CDNA5 / gfx1250, wave32. One thread = one pixel; one wave = 32 pixels.
// Phase 1: params = W.x + b via V_WMMA_F32_16X16X4_F32 (K = {x0,x1,x2,1}),
//          7 M-tiles x 2 N-tiles per wave, D tiles -> LDS.
// Phase 2: per-pixel censored-MoL math, log-free softmax chain.

typedef float v2f __attribute__((ext_vector_type(2)));
typedef float v8f __attribute__((ext_vector_type(8)));

#define NPARAM   100      // 10*M
#define PSTRIDE  116      // padded param stride (words): mult of 4 (b128 align), 52 mod 64 banks
#define WAVES    4
#define BLOCK    (WAVES * 32)
#define CH_STRIDE     4096      // H*W
#define BATCH_STRIDE  12288     // 3*H*W

__device__ __forceinline__ float fexp(float x) {
    return __builtin_amdgcn_exp2f(x * 1.4426950408889634f);   // v_exp_f32
}
__device__ __forceinline__ float frcp(float x) {
    return __builtin_amdgcn_rcpf(x);                          // v_rcp_f32
}
__device__ __forceinline__ float fsigmoid(float t) {
    return frcp(1.0f + fexp(-t));
}
__device__ __forceinline__ float ftanh(float y) {
    return 1.0f - 2.0f * frcp(fexp(2.0f * y) + 1.0f);
}

__global__ __launch_bounds__(BLOCK) void cmol_wmma_kernel(
    const float* __restrict__ xlo, const float* __restrict__ xup,
    const float* __restrict__ Wm,  const float* __restrict__ bvec,
    float* __restrict__ out, int zElems /* = B*3*H*W */)
{
    __shared__ __align__(16) float lds[WAVES * 32 * PSTRIDE];

    const int tid   = threadIdx.x;
    const int lane  = tid & 31;
    const int wave  = tid >> 5;
    const int laneL = lane & 15;
    const bool hi   = lane >= 16;

    const int q     = blockIdx.x * BLOCK + tid;   // flat pixel id over B*H*W
    const int bIdx  = q >> 12;                    // / (H*W)
    const int hw    = q & 4095;
    const int xbase = bIdx * BATCH_STRIDE + hw;

    // this lane's pixel: x_lower channels
    const float x0 = xlo[xbase];
    const float x1 = xlo[xbase + CH_STRIDE];
    const float x2 = xlo[xbase + 2 * CH_STRIDE];

    // ------------- Phase 1: GEMM via WMMA f32 16x16x4 -------------
    // A (16x4, f32 layout): lanes 0-15 hold K=0,1 ; lanes 16-31 hold K=2,3.
    // K rows = {W[:,0], W[:,1], W[:,2], bias} -> D = W.x + b in one pass.
    v2f afrag[7];
#pragma unroll
    for (int mt = 0; mt < 7; ++mt) {
        const int m = mt * 16 + laneL;
        float a0 = 0.0f, a1 = 0.0f;
        if (m < NPARAM) {
            if (hi) { a0 = Wm[m * 3 + 2]; a1 = bvec[m];      }
            else    { a0 = Wm[m * 3 + 0]; a1 = Wm[m * 3 + 1]; }
        }
        afrag[mt][0] = a0;
        afrag[mt][1] = a1;
    }

    float* myWave = &lds[wave * 32 * PSTRIDE];

#pragma unroll
    for (int t = 0; t < 2; ++t) {                 // two 16-pixel N-tiles per wave
        const int src = t * 16 + laneL;
        const float s0 = __shfl(x0, src, 32);
        const float s1 = __shfl(x1, src, 32);
        const float s2 = __shfl(x2, src, 32);
        // B (4x16): lanes 0-15 -> rows K=0,1 ; lanes 16-31 -> rows K=2,3 (col = lane%16)
        v2f bfrag;
        bfrag[0] = hi ? s2   : s0;
        bfrag[1] = hi ? 1.0f : s1;

#pragma unroll
        for (int mt = 0; mt < 7; ++mt) {
            v8f acc = {0.f, 0.f, 0.f, 0.f, 0.f, 0.f, 0.f, 0.f};
            v8f d = __builtin_amdgcn_wmma_f32_16x16x4_f32(
                false, afrag[mt], false, bfrag, (short)0, acc, false, false);
            // D layout: vgpr g, lanes 0-15 -> M=g ; lanes 16-31 -> M=8+g (N = lane%16)
            const int pidx = t * 16 + laneL;
            float* dst = &myWave[pidx * PSTRIDE + mt * 16 + (hi ? 8 : 0)];
            *(float4*)(dst)     = make_float4(d[0], d[1], d[2], d[3]);
            *(float4*)(dst + 4) = make_float4(d[4], d[5], d[6], d[7]);
        }
    }
    __syncthreads();   // order LDS stores vs per-lane reads

    // ------------- Phase 2: per-pixel censored-MoL math -------------
    const float* P = &myWave[lane * PSTRIDE];   // this lane's 100 params

    const float xu0 = xup[xbase];
    const float xu1 = xup[xbase + CH_STRIDE];
    const float xu2 = xup[xbase + 2 * CH_STRIDE];

    const float xlArr[3] = {x0, x1, x2};
    const float xuArr[3] = {xu0, xu1, xu2};
    const float xr0 = 2.0f * x0 - 1.0f;   // autoregressive terms use x_lower
    const float xr1 = 2.0f * x1 - 1.0f;

    // unnormalized mixture weights: exp(logit - max); softmax shift-invariance
    // collapses the whole log_softmax chain to running products with log_bin.
    float wgt[10];
    {
        float mx = P[0];
#pragma unroll
        for (int m = 1; m < 10; ++m) mx = fmaxf(mx, P[m]);
#pragma unroll
        for (int m = 0; m < 10; ++m) wgt[m] = fexp(P[m] - mx);
    }

#pragma unroll
    for (int c = 0; c < 3; ++c) {
        const float xL  = xlArr[c], xU = xuArr[c];
        const float xrL = 2.0f * xL - 1.0f;
        const float xrU = 2.0f * xU - 1.0f;
        float zlo = 0.0f, zup = 0.0f, wsum = 0.0f;
        float dbin[10];
#pragma unroll
        for (int m = 0; m < 10; ++m) {
            float mean = P[10 + c * 10 + m];
            if (c == 1) mean += ftanh(P[70 + m]) * xr0;
            if (c == 2) mean += ftanh(P[80 + m]) * xr0 + ftanh(P[90 + m]) * xr1;
            const float ls    = fmaxf(P[40 + c * 10 + m], -7.0f);
            const float inv_s = fexp(-ls);
            const float clo = (xL <= 0.0f) ? 0.0f : fsigmoid((xrL - mean) * inv_s);
            const float cup = (xU >= 1.0f) ? 1.0f : fsigmoid((xrU - mean) * inv_s);
            const float wm = wgt[m];
            zlo  += wm * clo;
            zup  += wm * cup;
            wsum += wm;
            dbin[m] = fmaxf(cup - clo, 1e-12f);
        }
        const float inv = frcp(wsum);
        zlo = fminf(fmaxf(zlo * inv, 0.0f), 1.0f);
        zup = fminf(fmaxf(zup * inv, 0.0f), 1.0f);

        const int oidx = bIdx * BATCH_STRIDE + c * CH_STRIDE + hw;
        out[oidx]          = zlo;
        out[zElems + oidx] = zup;

        if (c < 2) {
#pragma unroll
            for (int m = 0; m < 10; ++m) wgt[m] *= dbin[m];
        }
    }
}

extern "C" void kernel_launch(void* const* d_in, const int* in_sizes, int n_in,
                              void* d_out, int out_size, void* d_ws, size_t ws_size,
                              hipStream_t stream) {
    const float* xlo = (const float*)d_in[0];
    const float* xup = (const float*)d_in[1];
    const float* Wm  = (const float*)d_in[2];
    const float* bv  = (const float*)d_in[3];
    float* out = (float*)d_out;

    const int zElems = in_sizes[0];        // B*3*H*W
    const int npix   = zElems / 3;         // B*H*W = 262144 (exact multiple of BLOCK)
    const int blocks = npix / BLOCK;

    cmol_wmma_kernel<<<blocks, BLOCK, 0, stream>>>(xlo, xup, Wm, bv, out, zElems);
}